// DeChunkLayer_39522289058436
// MI455X (gfx1250) — compile-verified
//
#include <hip/hip_runtime.h>

#define B_    8
#define SEQ_  4096
#define MAXC_ 2048
#define DIM_  1024

typedef __attribute__((ext_vector_type(2))) float v2f;
typedef __attribute__((ext_vector_type(8))) float v8f;

// ---------------------------------------------------------------------------
// Kernel 1: per-batch boundary scan.
// Produces:
//   gates[b][c]  : clipped p at sorted rank c (boundary ranks first, then
//                  non-boundary ranks — matches reference's argsort gather)
//   pos[b][c]    : token index of the c-th boundary (c < nb)
//   cids[b][s]   : chunk id of token s (cumsum(mask)-1, clamped)
//   nbArr[b]     : min(#boundaries, MAXC)
// ---------------------------------------------------------------------------
__global__ void __launch_bounds__(1024)
dechunk_setup_kernel(const unsigned char* __restrict__ mask,
                     const float* __restrict__ prob,
                     float* __restrict__ gates,
                     int* __restrict__ pos,
                     int* __restrict__ cids,
                     int* __restrict__ nbArr)
{
    __shared__ int sc[1024];
    const int b = blockIdx.x;
    const int t = threadIdx.x;

    // 4 consecutive mask bytes per thread (coalesced u32 load)
    const unsigned int mw =
        ((const unsigned int*)(mask + (size_t)b * SEQ_))[t];
    const int m0 = (mw >>  0) & 1;
    const int m1 = (mw >>  8) & 1;
    const int m2 = (mw >> 16) & 1;
    const int m3 = (mw >> 24) & 1;
    const int cnt = m0 + m1 + m2 + m3;

    sc[t] = cnt;
    __syncthreads();
    // Hillis–Steele inclusive scan over 1024 threads
    #pragma unroll 1
    for (int off = 1; off < 1024; off <<= 1) {
        int v = (t >= off) ? sc[t - off] : 0;
        __syncthreads();
        sc[t] += v;
        __syncthreads();
    }
    const int incl  = sc[t];
    const int excl  = incl - cnt;
    const int total = sc[1023];

    float* gb = gates + (size_t)b * MAXC_;
    int*   pb = pos   + (size_t)b * MAXC_;
    int*   cb = cids  + (size_t)b * SEQ_;

    int bprefix = excl;  // boundaries strictly before current token
    const int ms[4] = {m0, m1, m2, m3};
    #pragma unroll
    for (int j = 0; j < 4; ++j) {
        const int s = t * 4 + j;
        const int m = ms[j];
        float p = prob[((size_t)b * SEQ_ + s) * 2 + 1];
        p = fminf(fmaxf(p, 1e-4f), 1.0f - 1e-4f);

        int cid = bprefix + m - 1;
        cid = (cid < 0) ? 0 : ((cid > MAXC_ - 1) ? MAXC_ - 1 : cid);
        cb[s] = cid;

        if (m) {
            if (bprefix < MAXC_) { pb[bprefix] = s; gb[bprefix] = p; }
            ++bprefix;
        } else {
            const int r = total + (s - bprefix);  // non-boundary sorted rank
            if (r < MAXC_) gb[r] = p;
        }
    }
    if (t == 0) nbArr[b] = (total < MAXC_) ? total : MAXC_;
}

// ---------------------------------------------------------------------------
// Kernel 2: fused blocked EMA scan (WMMA f32 16x16x4) + token scatter.
// Grid: (DIM_/128 slices, B). Block: 256 threads = 8 waves, one 16-channel
// N-tile per wave. Sequential over blocks of 16 chunks; carry in registers.
// ---------------------------------------------------------------------------
__global__ void __launch_bounds__(256)
dechunk_scan_kernel(const float* __restrict__ X,      // [B, MAXC, DIM]
                    const float* __restrict__ gates,  // [B, MAXC]
                    const int* __restrict__ pos,      // [B, MAXC]
                    const int* __restrict__ cids,     // [B, SEQ]
                    const int* __restrict__ nbArr,    // [B]
                    float* __restrict__ out)          // [B, SEQ, DIM]
{
    __shared__ float sg[16];
    __shared__ float sW[16][17];   // W[c][j], padded stride (bank-friendly)
    __shared__ float sP[16];       // P[c] = prod_{k<=c}(1-g_k)
    __shared__ __align__(16) float hblk[16][128];

    const int b         = blockIdx.y;
    const int sliceBase = blockIdx.x * 128;
    const int tid  = threadIdx.x;
    const int wave = tid >> 5;
    const int lane = tid & 31;
    const int hi   = lane >> 4;    // lane-half selects K pair / M half
    const int lo   = lane & 15;
    const int n    = sliceBase + wave * 16 + lo;   // this lane's channel

    const float* Xb   = X     + (size_t)b * MAXC_ * DIM_;
    const float* gb   = gates + (size_t)b * MAXC_;
    const int*   posb = pos   + (size_t)b * MAXC_;
    const int*   cidb = cids  + (size_t)b * SEQ_;

    const int nbc  = nbArr[b];
    const int nblk = (nbc + 15) >> 4;

    float hin = 0.0f;   // carry h_{block_in}[n], duplicated in both lane halves

    #pragma unroll 1
    for (int blk = 0; blk < nblk; ++blk) {
        const int cbase = blk << 4;

        __syncthreads();  // previous iteration's hblk readers done
        if (tid < 16) sg[tid] = gb[cbase + tid];
        __syncthreads();

        {   // thread t -> entry (c = t/16, j = t%16) of W; column 0 also does P
            const int c = tid >> 4, j = tid & 15;
            float w = 0.0f;
            if (j <= c) {
                w = sg[j];
                for (int k = j + 1; k <= c; ++k) w *= (1.0f - sg[k]);
            }
            sW[c][j] = w;
            if (j == 0) {
                float pp = 1.0f;
                for (int k = 0; k <= c; ++k) pp *= (1.0f - sg[k]);
                sP[c] = pp;
            }
        }
        __syncthreads();

        // C init: D[M][N] starts as P[M] * h_in[N]
        v8f acc;
        #pragma unroll
        for (int r = 0; r < 8; ++r) acc[r] = sP[r + 8 * hi] * hin;

        // D += W(16x16) @ X(16x16) as 4 fp32 WMMAs of K=4
        #pragma unroll
        for (int s = 0; s < 4; ++s) {
            const int k0 = 4 * s + 2 * hi;
            v2f a, bv;
            a.x = sW[lo][k0];
            a.y = sW[lo][k0 + 1];
            const float* xp = Xb + (size_t)(cbase + k0) * DIM_ + n;
            bv.x = xp[0];
            bv.y = xp[DIM_];
            acc = __builtin_amdgcn_wmma_f32_16x16x4_f32(
                false, a, false, bv, (short)0, acc, false, false);
        }

        // carry for next block: h[15][n] lives in reg 7 of lanes 16..31
        const float hnext = __shfl(acc[7], 16 + lo, 32);

        // stage this wave's 16x16 tile into the 16x128 block buffer
        #pragma unroll
        for (int r = 0; r < 8; ++r)
            hblk[r + 8 * hi][wave * 16 + lo] = acc[r];
        hin = hnext;
        __syncthreads();

        // dechunk scatter: tokens of this block are contiguous
        const int tokS = posb[cbase];
        const int tokE = (cbase + 16 < nbc) ? posb[cbase + 16] : SEQ_;
        for (int t = tokS + wave; t < tokE; t += 8) {
            int cl = cidb[t] - cbase;
            cl = (cl < 0) ? 0 : ((cl > 15) ? 15 : cl);
            const float4 v = *((const float4*)&hblk[cl][0] + lane);
            *((float4*)(out + (size_t)(b * SEQ_ + t) * DIM_ + sliceBase) + lane) = v;
        }
    }
}

// ---------------------------------------------------------------------------
extern "C" void kernel_launch(void* const* d_in, const int* in_sizes, int n_in,
                              void* d_out, int out_size, void* d_ws, size_t ws_size,
                              hipStream_t stream) {
    (void)in_sizes; (void)n_in; (void)out_size; (void)ws_size;
    const float*         chunk_states  = (const float*)d_in[0];
    const unsigned char* boundary_mask = (const unsigned char*)d_in[1];  // numpy bool = 1B
    const float*         boundary_prob = (const float*)d_in[2];
    float* out = (float*)d_out;

    char* ws = (char*)d_ws;
    float* gates = (float*)ws;                                        // B*MAXC f32
    int*   pos   = (int*)(ws + (size_t)B_ * MAXC_ * 4);               // B*MAXC i32
    int*   cids  = (int*)(ws + (size_t)2 * B_ * MAXC_ * 4);           // B*SEQ  i32
    int*   nbA   = (int*)(ws + (size_t)2 * B_ * MAXC_ * 4
                             + (size_t)B_ * SEQ_ * 4);                // B i32

    dechunk_setup_kernel<<<B_, 1024, 0, stream>>>(
        boundary_mask, boundary_prob, gates, pos, cids, nbA);

    dechunk_scan_kernel<<<dim3(DIM_ / 128, B_), 256, 0, stream>>>(
        chunk_states, gates, pos, cids, nbA, out);
}